// Detector_60670708023489
// MI455X (gfx1250) — compile-verified
//
#include <hip/hip_runtime.h>
#include <hip/hip_bf16.h>

// ---------------------------------------------------------------------------
// RetinaNet postprocess for MI455X (gfx1250, wave32).
//
// Phases:
//  k0: zero histograms/params                          (trivial)
//  k1: per-anchor max-logit/label, sigmoid, 4096-bin score histogram
//      -> HBM-bound: 96MB logits @23.3TB/s ~= 5us. float2 loads + prefetch.
//  k2: per-image threshold bin via suffix-sum over histogram.
//      Suffix-sum of 256 partials done with V_WMMA_F32_16X16X4_F32:
//      suffix = P * M, M[k][c] = (k>=c) triangular ones.  (CDNA5 matrix path)
//  k3: compact all candidates with bin >= cut (capacity 4096)
//  k4: per image (1 workgroup, 1024 thr = 32 waves):
//      bitonic sort 4096 keys desc by (score, index)  -> exact, deterministic
//      top-1000 -> box decode -> 1000-iter gaussian soft-NMS (wave32 shfl
//      argmax reductions) -> stable partition -> 1000-iter greedy IoU>0.8 ->
//      first 300 survivors written out.
// ---------------------------------------------------------------------------

typedef __attribute__((ext_vector_type(2))) float v2f;
typedef __attribute__((ext_vector_type(8))) float v8f;

#define B_    16
#define N_    150000
#define NBINS 4096
#define KTOP  1000
#define CAP   4096
#define DETS  300

__device__ __forceinline__ unsigned sortableF(float f) {
    unsigned u = __float_as_uint(f);
    return u ^ ((unsigned)((int)u >> 31) | 0x80000000u);
}
__device__ __forceinline__ float unsortableF(unsigned s) {
    unsigned u = (s & 0x80000000u) ? (s ^ 0x80000000u) : ~s;
    return __uint_as_float(u);
}
__device__ __forceinline__ float iouF(float ax1, float ay1, float ax2, float ay2,
                                      float bx1, float by1, float bx2, float by2) {
    float lx = fmaxf(ax1, bx1), ly = fmaxf(ay1, by1);
    float rx = fminf(ax2, bx2), ry = fminf(ay2, by2);
    float w = fmaxf(rx - lx, 0.0f), h = fmaxf(ry - ly, 0.0f);
    float inter = w * h;
    float a1 = (ax2 - ax1) * (ay2 - ay1);
    float a2 = (bx2 - bx1) * (by2 - by1);
    return inter / (a1 + a2 - inter + 1e-9f);
}

// ---------------------------------------------------------------------------
__global__ __launch_bounds__(256) void k0_zero(unsigned* hist, int* params) {
    int i = blockIdx.x * 256 + threadIdx.x;
    if (i < B_ * NBINS) hist[i] = 0u;
    if (i < B_ * 8) params[i] = 0;
}

// ---------------------------------------------------------------------------
// k1: score + histogram.  grid = (74, B_), block = 256.
__global__ __launch_bounds__(256) void k1_score(const float* __restrict__ logits,
                                                float* __restrict__ scoreArr,
                                                unsigned char* __restrict__ labArr,
                                                unsigned* __restrict__ hist) {
    const int img = blockIdx.y;
    const int tid = threadIdx.x;
    __shared__ unsigned lhist[NBINS];
    for (int b = tid; b < NBINS; b += 256) lhist[b] = 0u;
    __syncthreads();

    const int stride = gridDim.x * 256;
    for (int i = blockIdx.x * 256 + tid; i < N_; i += stride) {
        const float2* lg = (const float2*)logits + ((size_t)img * N_ + i) * 5;
        __builtin_prefetch((const void*)(lg + (size_t)stride * 5), 0, 1);  // global_prefetch_b8
        float2 p0 = lg[0], p1 = lg[1], p2 = lg[2], p3 = lg[3], p4 = lg[4];
        float v[10] = {p0.x, p0.y, p1.x, p1.y, p2.x, p2.y, p3.x, p3.y, p4.x, p4.y};
        float m = v[0];
        int lbl = 0;
#pragma unroll
        for (int c = 1; c < 10; ++c)
            if (v[c] > m) { m = v[c]; lbl = c; }   // first-occurrence argmax
        float sc = 1.0f / (1.0f + __expf(-m));     // sigmoid(max) == max(sigmoid)
        size_t o = (size_t)img * N_ + i;
        if (sc > 0.05f) {
            scoreArr[o] = sc;
            labArr[o] = (unsigned char)lbl;
            int bin = (int)(sc * (float)NBINS);
            bin = bin > NBINS - 1 ? NBINS - 1 : bin;
            atomicAdd(&lhist[bin], 1u);
        } else {
            scoreArr[o] = -1.0f;                   // fails score_thresh
        }
    }
    __syncthreads();
    for (int b = tid; b < NBINS; b += 256) {
        unsigned c = lhist[b];
        if (c) atomicAdd(&hist[img * NBINS + b], c);
    }
}

// ---------------------------------------------------------------------------
// k2: per-image cut-bin via WMMA suffix sum.  grid = B_, block = 256.
// params[img*8+0] = cut bin, params[img*8+1] = compaction counter (zeroed by k0)
__global__ __launch_bounds__(256) void k2_threshold(const unsigned* __restrict__ hist,
                                                    int* __restrict__ params,
                                                    unsigned long long* __restrict__ candKeys) {
    const int img = blockIdx.x;
    const int tid = threadIdx.x;
    __shared__ float partial[256];
    __shared__ float sfx[256];
    __shared__ float tail[16];
    __shared__ int found;

    // init candidate buffer for this image (padding key 0 sorts last)
    for (int i = tid; i < CAP; i += 256) candKeys[(size_t)img * CAP + i] = 0ull;
    if (tid == 0) found = 0;

    // each thread owns 16 bins; partial[t] = sum of its bins
    float h[16];
    float psum = 0.0f;
    const unsigned* hb = hist + img * NBINS + tid * 16;
#pragma unroll
    for (int j = 0; j < 16; ++j) { h[j] = (float)hb[j]; psum += h[j]; }
    partial[tid] = psum;
    __syncthreads();

    // Wave 0: suffix-sum of the 256 partials via matrix core.
    // View partials as P[16][16] row-major; suffix_incl = P*M, M[k][c]=(k>=c).
    // K=16 done as 4 chained V_WMMA_F32_16X16X4_F32 (f32 keeps counts exact).
    if (tid < 32) {
        const int lane = tid;
        const int half = (lane >= 16);          // lanes 16..31 carry K=2,3 / M rows +8
        const int m = lane & 15;
        v8f acc = {};
#pragma unroll
        for (int q = 0; q < 4; ++q) {
            // A (16x4) layout: lanes0-15 -> M=lane, VGPR0=K0,VGPR1=K1; lanes16-31 -> K2,K3
            const int k0 = 4 * q + (half ? 2 : 0);
            v2f a, b;
            a.x = partial[16 * m + k0];
            a.y = partial[16 * m + k0 + 1];
            // B (4x16): VGPR0 rows {k0}, VGPR1 rows {k0+1}; col = lane&15
            b.x = (k0 >= m) ? 1.0f : 0.0f;
            b.y = (k0 + 1 >= m) ? 1.0f : 0.0f;
            acc = __builtin_amdgcn_wmma_f32_16x16x4_f32(false, a, false, b,
                                                        (short)0, acc, false, false);
        }
        // D layout: VGPR g -> row (g + half*8), col (lane&15)
#pragma unroll
        for (int g = 0; g < 8; ++g) sfx[16 * (g + (half ? 8 : 0)) + m] = acc[g];
    }
    __syncthreads();
    // tail[r] = sum of full-row sums of rows > r; rowsum[k] = sfx[16k + 0]
    if (tid < 16) {
        float s = 0.0f;
        for (int k = tid + 1; k < 16; ++k) s += sfx[16 * k];
        tail[tid] = s;
    }
    __syncthreads();

    // S(bin) = count of scores with bin' >= bin.  Find cut: S(cut)>=K, S(cut+1)<K.
    float SPnext = (tid == 255) ? 0.0f : (sfx[tid + 1] + tail[(tid + 1) >> 4]);
    float ls[17];
    ls[16] = 0.0f;
#pragma unroll
    for (int j = 15; j >= 0; --j) ls[j] = ls[j + 1] + h[j];
#pragma unroll
    for (int j = 0; j < 16; ++j) {
        float Sb = ls[j] + SPnext;
        float Sb1 = ls[j + 1] + SPnext;
        if (Sb >= (float)KTOP && Sb1 < (float)KTOP) {
            params[img * 8 + 0] = 16 * tid + j;
            found = 1;
        }
    }
    __syncthreads();
    if (tid == 0 && !found) params[img * 8 + 0] = 0;  // total < K: take everything
}

// ---------------------------------------------------------------------------
// k3: compact candidates with bin >= cut.  grid = (74, B_), block = 256.
__global__ __launch_bounds__(256) void k3_compact(const float* __restrict__ scoreArr,
                                                  const unsigned char* __restrict__ labArr,
                                                  int* __restrict__ params,
                                                  unsigned long long* __restrict__ candKeys) {
    const int img = blockIdx.y;
    const int cut = params[img * 8 + 0];
    const int stride = gridDim.x * 256;
    for (int i = blockIdx.x * 256 + threadIdx.x; i < N_; i += stride) {
        float s = scoreArr[(size_t)img * N_ + i];
        if (s > 0.0f) {
            int bin = (int)(s * (float)NBINS);
            bin = bin > NBINS - 1 ? NBINS - 1 : bin;
            if (bin >= cut) {
                int p = atomicAdd(&params[img * 8 + 1], 1);
                if (p < CAP) {
                    unsigned lab = labArr[(size_t)img * N_ + i];
                    // tiebreak: larger key == smaller anchor index (top_k semantics)
                    unsigned low = ((0x3FFFFu - (unsigned)i) << 8) | lab;
                    candKeys[(size_t)img * CAP + p] =
                        ((unsigned long long)sortableF(s) << 32) | low;
                }
            }
        }
    }
}

// ---------------------------------------------------------------------------
struct K4Smem {
    union {
        unsigned long long keys[CAP];  // phase 1 only (32KB)
        struct {
            float bx1[1024], by1[1024], bx2[1024], by2[1024];
            float off[1024];
            float ssc[1024];
            int slab[1024];
            unsigned alive[1024];
            int scan[1024];
            unsigned long long wred[33];
        } p;
    };
};

__device__ __forceinline__ int blockScanExcl(int val, volatile int* buf, int tid, int* total) {
    buf[tid] = val;
    __syncthreads();
    for (int off = 1; off < 1024; off <<= 1) {
        int x = (tid >= off) ? buf[tid - off] : 0;
        __syncthreads();
        buf[tid] += x;
        __syncthreads();
    }
    int incl = buf[tid];
    *total = buf[1023];
    __syncthreads();
    return incl - val;
}

// k4: sort + soft-NMS + greedy NMS + output.  grid = B_, block = 1024.
__global__ __launch_bounds__(1024) void k4_nms(const float4* __restrict__ rel,
                                               const float4* __restrict__ anc,
                                               const unsigned long long* __restrict__ candKeys,
                                               float* __restrict__ out) {
    const int img = blockIdx.x;
    const int tid = threadIdx.x;
    const int lane = tid & 31;
    const int wid = tid >> 5;
    __shared__ K4Smem sm;

    // ---- phase 1: bitonic sort CAP keys descending ----
    for (int i = tid; i < CAP; i += 1024) sm.keys[i] = candKeys[(size_t)img * CAP + i];
    for (unsigned k = 2; k <= CAP; k <<= 1) {
        for (unsigned j = k >> 1; j; j >>= 1) {
            __syncthreads();
            for (unsigned i = tid; i < CAP; i += 1024) {
                unsigned ixj = i ^ j;
                if (ixj > i) {
                    unsigned long long a = sm.keys[i], b = sm.keys[ixj];
                    bool descDir = ((i & k) == 0);
                    bool sw = descDir ? (a < b) : (a > b);
                    if (sw) { sm.keys[i] = b; sm.keys[ixj] = a; }
                }
            }
        }
    }
    __syncthreads();
    unsigned long long myKey = (tid < KTOP) ? sm.keys[tid] : 0ull;
    __syncthreads();  // union reuse below

    const bool real = (myKey >> 32) != 0ull;
    float myScore = real ? unsortableF((unsigned)(myKey >> 32)) : -__builtin_inff();
    int myIdx = real ? (int)(0x3FFFFu - (((unsigned)myKey >> 8) & 0x3FFFFu)) : 0;
    int myLab = real ? (int)((unsigned)myKey & 0xFFu) : 0;

    // ---- box decode (torchvision BoxCoder, weights 1; clip to image) ----
    float b1 = 0.f, b2 = 0.f, b3 = 0.f, b4 = 0.f, myOff = 0.f;
    if (real) {
        float4 r = rel[(size_t)img * N_ + myIdx];
        float4 a = anc[(size_t)img * N_ + myIdx];
        float wa = a.z - a.x, ha = a.w - a.y;
        float cxa = a.x + 0.5f * wa, cya = a.y + 0.5f * ha;
        const float CLIP = 4.135166556742356f;  // log(1000/16)
        float dw = fminf(r.z, CLIP), dh = fminf(r.w, CLIP);
        float cx = r.x * wa + cxa, cy = r.y * ha + cya;
        float w = __expf(dw) * wa, h = __expf(dh) * ha;
        b1 = fminf(fmaxf(cx - 0.5f * w, 0.f), 800.f);
        b2 = fminf(fmaxf(cy - 0.5f * h, 0.f), 800.f);
        b3 = fminf(fmaxf(cx + 0.5f * w, 0.f), 800.f);
        b4 = fminf(fmaxf(cy + 0.5f * h, 0.f), 800.f);
        myOff = (float)myLab * 801.0f;  // per-class offset (IMG_W + 1)
    }
    sm.p.bx1[tid] = b1; sm.p.by1[tid] = b2; sm.p.bx2[tid] = b3; sm.p.by2[tid] = b4;
    sm.p.off[tid] = myOff;
    __syncthreads();

    // ---- phase 2: gaussian soft-NMS, 1000 serial argmax+decay steps ----
    float s = myScore;
    bool keepSoft = false;
    for (int it = 0; it < KTOP; ++it) {
        unsigned long long kk =
            ((unsigned long long)sortableF(s) << 32) | (unsigned)(1023 - tid);
#pragma unroll
        for (int m = 16; m >= 1; m >>= 1) {          // wave32 reduction
            unsigned long long o = __shfl_xor(kk, m, 32);
            if (o > kk) kk = o;
        }
        if (lane == 0) sm.p.wred[wid] = kk;
        __syncthreads();
        if (tid < 32) {
            unsigned long long k2 = sm.p.wred[tid];
#pragma unroll
            for (int m = 16; m >= 1; m >>= 1) {
                unsigned long long o = __shfl_xor(k2, m, 32);
                if (o > k2) k2 = o;
            }
            if (tid == 0) sm.p.wred[32] = k2;
        }
        __syncthreads();
        unsigned long long best = sm.p.wred[32];
        int sel = 1023 - (int)(best & 0xFFFFFFFFull);
        float val = unsortableF((unsigned)(best >> 32));
        float so = sm.p.off[sel];
        float iou = iouF(sm.p.bx1[sel] + so, sm.p.by1[sel] + so,
                         sm.p.bx2[sel] + so, sm.p.by2[sel] + so,
                         b1 + myOff, b2 + myOff, b3 + myOff, b4 + myOff);
        float decay = __expf(-(iou * iou) * 2.0f);   // 1/sigma = 2
        if (tid == sel) {
            s = -__builtin_inff();
            keepSoft = keepSoft || (val > 0.1f);
        } else {
            s *= decay;
        }
    }

    // ---- stable partition: survivors (score-desc order) first ----
    int total;
    int excl = blockScanExcl(keepSoft ? 1 : 0, sm.p.scan, tid, &total);
    int newpos = keepSoft ? excl : total + (tid - excl);
    sm.p.bx1[newpos] = b1; sm.p.by1[newpos] = b2;
    sm.p.bx2[newpos] = b3; sm.p.by2[newpos] = b4;
    sm.p.ssc[newpos] = myScore;
    sm.p.slab[newpos] = myLab;
    sm.p.alive[newpos] = keepSoft ? 1u : 0u;
    __syncthreads();

    float g1 = sm.p.bx1[tid], g2 = sm.p.by1[tid], g3 = sm.p.bx2[tid], g4 = sm.p.by2[tid];
    float gsc = sm.p.ssc[tid];
    int glab = sm.p.slab[tid];

    // ---- phase 3: greedy IoU>0.8 suppression, 1000 serial steps ----
    bool keepF = false;
    for (int i = 0; i < KTOP; ++i) {
        __syncthreads();
        if (sm.p.alive[i]) {
            if (tid == i) {
                keepF = true;
            } else if (sm.p.alive[tid]) {
                float iou = iouF(sm.p.bx1[i], sm.p.by1[i], sm.p.bx2[i], sm.p.by2[i],
                                 g1, g2, g3, g4);
                if (iou > 0.8f) sm.p.alive[tid] = 0u;
            }
        }
    }
    __syncthreads();

    // ---- output: first 300 survivors (fixed shape, padded) ----
    int tf;
    int rank = blockScanExcl(keepF ? 1 : 0, sm.p.scan, tid, &tf);
    float* outBoxes = out;
    float* outScores = out + (size_t)B_ * DETS * 4;
    float* outLabels = outScores + (size_t)B_ * DETS;
    float* outValid = outLabels + (size_t)B_ * DETS;
    if (keepF && rank < DETS) {
        size_t ob = ((size_t)img * DETS + rank) * 4;
        outBoxes[ob + 0] = g1; outBoxes[ob + 1] = g2;
        outBoxes[ob + 2] = g3; outBoxes[ob + 3] = g4;
        outScores[(size_t)img * DETS + rank] = gsc;
        outLabels[(size_t)img * DETS + rank] = (float)glab;
        outValid[(size_t)img * DETS + rank] = 1.0f;
    }
    if (tid < DETS && tid >= tf) {  // padding slots
        size_t ob = ((size_t)img * DETS + tid) * 4;
        outBoxes[ob + 0] = 0.f; outBoxes[ob + 1] = 0.f;
        outBoxes[ob + 2] = 0.f; outBoxes[ob + 3] = 0.f;
        outScores[(size_t)img * DETS + tid] = 0.f;
        outLabels[(size_t)img * DETS + tid] = -1.0f;
        outValid[(size_t)img * DETS + tid] = 0.f;
    }
}

// ---------------------------------------------------------------------------
extern "C" void kernel_launch(void* const* d_in, const int* in_sizes, int n_in,
                              void* d_out, int out_size, void* d_ws, size_t ws_size,
                              hipStream_t stream) {
    (void)in_sizes; (void)n_in; (void)out_size; (void)ws_size;
    const float* logits = (const float*)d_in[0];
    const float4* rel = (const float4*)d_in[1];
    const float4* anc = (const float4*)d_in[2];
    float* out = (float*)d_out;

    // workspace carve (all 256B aligned): hist | params | scores | labels | keys
    char* ws = (char*)d_ws;
    size_t o = 0;
    auto align256 = [&](size_t x) { return (x + 255) & ~(size_t)255; };
    unsigned* hist = (unsigned*)(ws + o);          o = align256(o + (size_t)B_ * NBINS * 4);
    int* params = (int*)(ws + o);                  o = align256(o + (size_t)B_ * 8 * 4);
    float* scoreArr = (float*)(ws + o);            o = align256(o + (size_t)B_ * N_ * 4);
    unsigned char* labArr = (unsigned char*)(ws + o); o = align256(o + (size_t)B_ * N_);
    unsigned long long* candKeys = (unsigned long long*)(ws + o);

    k0_zero<<<(B_ * NBINS + 255) / 256, 256, 0, stream>>>(hist, params);
    k1_score<<<dim3(74, B_), 256, 0, stream>>>(logits, scoreArr, labArr, hist);
    k2_threshold<<<B_, 256, 0, stream>>>(hist, params, candKeys);
    k3_compact<<<dim3(74, B_), 256, 0, stream>>>(scoreArr, labArr, params, candKeys);
    k4_nms<<<B_, 1024, 0, stream>>>(rel, anc, candKeys, out);
}